// SVD_88802743812379
// MI455X (gfx1250) — compile-verified
//
#include <hip/hip_runtime.h>

typedef __attribute__((ext_vector_type(2))) float v2f;
typedef __attribute__((ext_vector_type(4))) float v4f;
typedef __attribute__((ext_vector_type(8))) float v8f;

#define KB_B 8192
#define KB_N 2048

// ---------------------------------------------------------------------------
// Phase 1: bandwidth-bound streaming moment reduction.
// One block per batch. 256 threads * 24 floats (8 points) = 6144 = N*3 floats.
// Accumulates sx,sy,sz and the 6 unique entries of sum(x x^T).
// ---------------------------------------------------------------------------
__global__ __launch_bounds__(256)
void kabsch_moments(const float* __restrict__ src, float* __restrict__ ws)
{
    const int b   = blockIdx.x;
    const int tid = threadIdx.x;
    const v4f* p =
        reinterpret_cast<const v4f*>(src + (size_t)b * (KB_N * 3) + (size_t)tid * 24);

    float buf[24];
#pragma unroll
    for (int i = 0; i < 6; ++i) {
        v4f v = __builtin_nontemporal_load(&p[i]);   // streaming, bypass-temporal
        buf[4*i+0] = v.x; buf[4*i+1] = v.y; buf[4*i+2] = v.z; buf[4*i+3] = v.w;
    }

    float acc[9] = {0,0,0,0,0,0,0,0,0};   // sx sy sz xx yy zz xy xz yz
#pragma unroll
    for (int k = 0; k < 8; ++k) {
        float x = buf[3*k+0], y = buf[3*k+1], z = buf[3*k+2];
        acc[0] += x; acc[1] += y; acc[2] += z;
        acc[3] = fmaf(x, x, acc[3]);
        acc[4] = fmaf(y, y, acc[4]);
        acc[5] = fmaf(z, z, acc[5]);
        acc[6] = fmaf(x, y, acc[6]);
        acc[7] = fmaf(x, z, acc[7]);
        acc[8] = fmaf(y, z, acc[8]);
    }

    // wave32 butterfly reduction
#pragma unroll
    for (int k = 0; k < 9; ++k) {
        float v = acc[k];
#pragma unroll
        for (int off = 16; off > 0; off >>= 1)
            v += __shfl_down(v, off, 32);
        acc[k] = v;
    }

    __shared__ float part[8][9];
    const int wave = tid >> 5, lane = tid & 31;
    if (lane == 0) {
#pragma unroll
        for (int k = 0; k < 9; ++k) part[wave][k] = acc[k];
    }
    __syncthreads();
    if (tid < 9) {
        float t = 0.f;
#pragma unroll
        for (int w = 0; w < 8; ++w) t += part[w][tid];
        ws[(size_t)b * 12 + tid] = t;
    }
}

// ---------------------------------------------------------------------------
// 3x3 symmetric Jacobi eigendecomposition (lane-uniform, branch-light).
// A is overwritten; on exit diag(A) = eigenvalues, V columns = eigenvectors.
// ---------------------------------------------------------------------------
__device__ __forceinline__ void eig3_sym(float A[3][3], float V[3][3])
{
    V[0][0]=1.f; V[0][1]=0.f; V[0][2]=0.f;
    V[1][0]=0.f; V[1][1]=1.f; V[1][2]=0.f;
    V[2][0]=0.f; V[2][1]=0.f; V[2][2]=1.f;
#pragma unroll
    for (int sweep = 0; sweep < 8; ++sweep) {
#pragma unroll
        for (int r = 0; r < 3; ++r) {
            const int p = (r == 2) ? 1 : 0;
            const int q = (r == 0) ? 1 : 2;
            float apq = A[p][q], app = A[p][p], aqq = A[q][q];
            bool  rot = fabsf(apq) > 1e-9f * (fabsf(app) + fabsf(aqq)) + 1e-32f;
            float den = 2.0f * apq + (rot ? 0.f : 1.f);
            float tau = (aqq - app) / den;
            float t   = copysignf(1.0f, tau) / (fabsf(tau) + sqrtf(1.0f + tau * tau));
            float cc  = 1.0f / sqrtf(1.0f + t * t);
            float ssn = t * cc;
            cc  = rot ? cc  : 1.0f;
            ssn = rot ? ssn : 0.0f;
#pragma unroll
            for (int i = 0; i < 3; ++i) {
                float aip = A[i][p], aiq = A[i][q];
                A[i][p] = cc * aip - ssn * aiq;
                A[i][q] = ssn * aip + cc * aiq;
            }
#pragma unroll
            for (int i = 0; i < 3; ++i) {
                float api = A[p][i], aqi = A[q][i];
                A[p][i] = cc * api - ssn * aqi;
                A[q][i] = ssn * api + cc * aqi;
            }
#pragma unroll
            for (int i = 0; i < 3; ++i) {
                float vip = V[i][p], viq = V[i][q];
                V[i][p] = cc * vip - ssn * viq;
                V[i][q] = ssn * vip + cc * viq;
            }
        }
    }
}

// ---------------------------------------------------------------------------
// Phase 2: one wave32 per batch. H = M2 - N*mu*mu^T; H symmetric PSD so its
// SVD is the eigendecomposition (U == V). R = V * U^T done with
// v_wmma_f32_16x16x4_f32 (3x3 live in the 16x16 tile, K padded 3->4).
// ---------------------------------------------------------------------------
__global__ __launch_bounds__(256)
void kabsch_finalize(const float* __restrict__ ws, float* __restrict__ out)
{
    const int lane = threadIdx.x & 31;
    const int b    = blockIdx.x * 8 + (threadIdx.x >> 5);
    const float* s = ws + (size_t)b * 12;
    const float invn = 1.0f / (float)KB_N;

    float sx = s[0], sy = s[1], sz = s[2];
    float mu[3] = { sx * invn, sy * invn, sz * invn };

    float A[3][3];
    A[0][0] = s[3] - sx * mu[0];
    A[1][1] = s[4] - sy * mu[1];
    A[2][2] = s[5] - sz * mu[2];
    A[0][1] = A[1][0] = s[6] - sx * mu[1];
    A[0][2] = A[2][0] = s[7] - sx * mu[2];
    A[1][2] = A[2][1] = s[8] - sy * mu[2];

    float V[3][3];
    eig3_sym(A, V);

    // sort eigenpairs descending (SVD convention)
    float w[3] = { A[0][0], A[1][1], A[2][2] };
#pragma unroll
    for (int a = 0; a < 2; ++a)
#pragma unroll
        for (int bb = 0; bb < 2 - a; ++bb) {
            bool sw = w[bb] < w[bb + 1];
            float tw = w[bb]; w[bb] = sw ? w[bb+1] : w[bb]; w[bb+1] = sw ? tw : w[bb+1];
#pragma unroll
            for (int i = 0; i < 3; ++i) {
                float tv = V[i][bb];
                V[i][bb]   = sw ? V[i][bb+1] : V[i][bb];
                V[i][bb+1] = sw ? tv         : V[i][bb+1];
            }
        }

    // Build WMMA fragments: A-frag = V (16x4, rows/cols >=3 zero),
    // B-frag = U^T with U == V  =>  B[k][n] = V[n][k].
    // f32 16x4 A layout: v0: lanes0-15 K=0, lanes16-31 K=2; v1: K=1 / K=3.
    const int  m  = lane & 15;
    const bool hi = lane >= 16;
    float a0 = 0.f, a1 = 0.f, b0 = 0.f, b1 = 0.f;
    if (m < 3) {
        a0 = hi ? V[m][2] : V[m][0];
        a1 = hi ? 0.0f    : V[m][1];
        b0 = hi ? V[m][2] : V[m][0];   // U == V
        b1 = hi ? 0.0f    : V[m][1];
    }
    v2f av; av.x = a0; av.y = a1;
    v2f bv; bv.x = b0; bv.y = b1;
    v8f c = {0.f,0.f,0.f,0.f,0.f,0.f,0.f,0.f};
    c = __builtin_amdgcn_wmma_f32_16x16x4_f32(
            /*neg_a=*/false, av, /*neg_b=*/false, bv,
            /*c_mod=*/(short)0, c, /*reuse_a=*/false, /*reuse_b=*/false);

    // Gather the 3x3 result to all lanes (D layout: c[i] lane n => R[i][n]).
    float R[3][3];
#pragma unroll
    for (int i = 0; i < 3; ++i)
#pragma unroll
        for (int j = 0; j < 3; ++j)
            R[i][j] = __shfl(c[i], j, 32);

    float det = R[0][0]*(R[1][1]*R[2][2]-R[1][2]*R[2][1])
              - R[0][1]*(R[1][0]*R[2][2]-R[1][2]*R[2][0])
              + R[0][2]*(R[1][0]*R[2][1]-R[1][1]*R[2][0]);

    if (det < 0.0f) {  // faithful to reference: r2 = ((V*refl)U^T) .* refl (elementwise)
        float R2[3][3];
#pragma unroll
        for (int i = 0; i < 3; ++i)
#pragma unroll
            for (int j = 0; j < 3; ++j) {
                float raw = V[i][0]*V[j][0] + V[i][1]*V[j][1] - V[i][2]*V[j][2];
                float refl = (i == j) ? ((i == 2) ? -1.f : 1.f) : 0.f;
                R2[i][j] = raw * refl;
            }
#pragma unroll
        for (int i = 0; i < 3; ++i)
#pragma unroll
            for (int j = 0; j < 3; ++j) R[i][j] = R2[i][j];
    }

    if (lane == 0) {
        float* Ro = out + (size_t)b * 9;
#pragma unroll
        for (int i = 0; i < 3; ++i)
#pragma unroll
            for (int j = 0; j < 3; ++j) Ro[i*3 + j] = R[i][j];
        float* To = out + (size_t)KB_B * 9 + (size_t)b * 3;
#pragma unroll
        for (int i = 0; i < 3; ++i)
            To[i] = mu[i] - (R[i][0]*mu[0] + R[i][1]*mu[1] + R[i][2]*mu[2]);
    }
}

// ---------------------------------------------------------------------------
extern "C" void kernel_launch(void* const* d_in, const int* in_sizes, int n_in,
                              void* d_out, int out_size, void* d_ws, size_t ws_size,
                              hipStream_t stream)
{
    const float* src = (const float*)d_in[0];   // source [8192,2048,3] f32
    // d_in[1] (template) is unused by the reference math.
    float* out = (float*)d_out;                 // R flat (B*9) then t flat (B*3)
    float* ws  = (float*)d_ws;                  // needs KB_B*12 floats = 384 KiB

    kabsch_moments <<<KB_B,     256, 0, stream>>>(src, ws);
    kabsch_finalize<<<KB_B / 8, 256, 0, stream>>>(ws, out);
}